// HistogramObserver_78323023610526
// MI455X (gfx1250) — compile-verified
//
#include <hip/hip_runtime.h>
#include <math.h>
#include <stdint.h>

#define BINS 2048
#define DST_NBINS 8
#define STEPF 1e-5f

typedef __attribute__((ext_vector_type(2))) float v2f;
typedef __attribute__((ext_vector_type(8))) float v8f;
typedef int v4i_gnu __attribute__((vector_size(16)));  // matches builtin's V4i

// ---- CDNA5 feature probes (compile-safe) ------------------------------------
#if defined(__AMDGCN__) && __has_builtin(__builtin_amdgcn_global_load_async_to_lds_b128) && __has_builtin(__builtin_amdgcn_s_wait_asynccnt)
#define USE_ASYNC 1
#endif
#if defined(__AMDGCN__) && __has_builtin(__builtin_amdgcn_wmma_f32_16x16x4_f32)
#define USE_WMMA 1
#endif

#ifdef USE_ASYNC
typedef __attribute__((address_space(1))) v4i_gnu* gv4ip;  // global src
typedef __attribute__((address_space(3))) v4i_gnu* lv4ip;  // LDS dst
__device__ __forceinline__ void async_copy16(const float* g, float* l) {
  // 16B per lane, global -> LDS, tracked by ASYNCcnt (bypasses VGPRs)
  __builtin_amdgcn_global_load_async_to_lds_b128(
      (gv4ip)(uintptr_t)g, (lv4ip)(unsigned)(uintptr_t)l, 0, 0);
}
#endif

// ---- float atomic min/max via monotonic int punning -------------------------
__device__ __forceinline__ void atomic_min_f32(float* addr, float val) {
  if (val >= 0.0f) atomicMin((int*)addr, __float_as_int(val));
  else             atomicMax((unsigned int*)addr, __float_as_uint(val));
}
__device__ __forceinline__ void atomic_max_f32(float* addr, float val) {
  if (val >= 0.0f) atomicMax((int*)addr, __float_as_int(val));
  else             atomicMin((unsigned int*)addr, __float_as_uint(val));
}

// ---- init: ws[0]=+inf(min), ws[1]=-inf(max), ws[2..2049]=hist zeros ---------
__global__ void init_kernel(float* ws) {
  int i = blockIdx.x * blockDim.x + threadIdx.x;
  if (i == 0) {
    ((unsigned*)ws)[0] = 0x7F800000u;  // +inf
    ((unsigned*)ws)[1] = 0xFF800000u;  // -inf
  }
  for (int j = i; j < BINS; j += gridDim.x * blockDim.x) ws[2 + j] = 0.0f;
}

// ---- pass 1: min/max (streams 128MB from HBM, leaves it in 192MB L2) --------
__global__ void __launch_bounds__(256) minmax_kernel(const float* __restrict__ x,
                                                     int n, float* __restrict__ ws) {
  float vmin =  __builtin_inff();
  float vmax = -__builtin_inff();
  const int n4 = n >> 2;
  const float4* x4 = (const float4*)x;
  const int stride = gridDim.x * blockDim.x;
  for (int i = blockIdx.x * blockDim.x + threadIdx.x; i < n4; i += stride) {
    float4 v = x4[i];  // global_load_b128
    vmin = fminf(vmin, fminf(fminf(v.x, v.y), fminf(v.z, v.w)));
    vmax = fmaxf(vmax, fmaxf(fmaxf(v.x, v.y), fmaxf(v.z, v.w)));
  }
  if (blockIdx.x == 0 && (int)threadIdx.x < (n & 3)) {  // scalar tail
    float v = x[(n4 << 2) + threadIdx.x];
    vmin = fminf(vmin, v); vmax = fmaxf(vmax, v);
  }
  for (int d = 16; d > 0; d >>= 1) {  // wave32 reduce
    vmin = fminf(vmin, __shfl_xor(vmin, d, 32));
    vmax = fmaxf(vmax, __shfl_xor(vmax, d, 32));
  }
  __shared__ float smin[8], smax[8];
  const int wave = threadIdx.x >> 5, lane = threadIdx.x & 31;
  if (lane == 0) { smin[wave] = vmin; smax[wave] = vmax; }
  __syncthreads();
  if (threadIdx.x == 0) {
    float m = smin[0], M = smax[0];
    const int nw = blockDim.x >> 5;
    for (int w = 1; w < nw; ++w) { m = fminf(m, smin[w]); M = fmaxf(M, smax[w]); }
    atomic_min_f32(&ws[0], m);
    atomic_max_f32(&ws[1], M);
  }
}

// ---- pass 2: LDS-privatized histogram, async double-buffered staging --------
__device__ __forceinline__ void bin1(float v, float mn, float inv, unsigned* bins) {
  int idx = (int)floorf((v - mn) * inv);
  idx = idx < 0 ? 0 : (idx > BINS - 1 ? BINS - 1 : idx);
  atomicAdd(&bins[idx], 1u);  // ds_add_u32
}
__device__ __forceinline__ void bin4(float4 v, float mn, float inv, unsigned* bins) {
  bin1(v.x, mn, inv, bins); bin1(v.y, mn, inv, bins);
  bin1(v.z, mn, inv, bins); bin1(v.w, mn, inv, bins);
}

__global__ void __launch_bounds__(256) hist_kernel(const float* __restrict__ x, int n,
                                                   const float* __restrict__ mm,
                                                   float* __restrict__ ghist) {
  __shared__ unsigned bins[BINS];
#ifdef USE_ASYNC
  __shared__ __align__(16) float stage[2][256 * 4];
#endif
  const int tid = threadIdx.x;
  for (int i = tid; i < BINS; i += blockDim.x) bins[i] = 0u;
  __syncthreads();

  const float mn = mm[0];
  const float mx = mm[1];
  const float bw = (mx - mn) / (float)BINS;
  const float inv = 1.0f / ((bw == 0.0f) ? 1.0f : bw);
  const int n4 = n >> 2;
  const int ntiles = n4 >> 8;  // tiles of 256 lanes x float4 = 4KB

#ifdef USE_ASYNC
  int t = (int)blockIdx.x;
  int buf = 0;
  if (t < ntiles)
    async_copy16(x + (((size_t)t << 8) + tid) * 4, &stage[0][tid << 2]);
  for (; t < ntiles; t += (int)gridDim.x) {
    const int tn = t + (int)gridDim.x;
    if (tn < ntiles) {  // uniform branch
      async_copy16(x + (((size_t)tn << 8) + tid) * 4, &stage[buf ^ 1][tid << 2]);
      __builtin_amdgcn_s_wait_asynccnt(1);  // oldest (current buf) complete
    } else {
      __builtin_amdgcn_s_wait_asynccnt(0);
    }
    const float4 v = *(const float4*)&stage[buf][tid << 2];  // own slot only
    bin4(v, mn, inv, bins);
    buf ^= 1;
  }
#else
  const float4* x4 = (const float4*)x;
  const int nvec = ntiles << 8;
  for (int i = (int)blockIdx.x * (int)blockDim.x + tid; i < nvec;
       i += (int)gridDim.x * (int)blockDim.x) {
    bin4(x4[i], mn, inv, bins);
  }
#endif
  if (blockIdx.x == 0) {  // leftover float4s + scalar tail
    for (int i = (ntiles << 10) + tid; i < n; i += blockDim.x)
      bin1(x[i], mn, inv, bins);
  }
  __syncthreads();
  for (int i = tid; i < BINS; i += blockDim.x) {
    const unsigned c = bins[i];
    if (c) atomicAdd(&ghist[i], (float)c);  // global_atomic_add_f32
  }
}

// ---- wave32 sum: V_WMMA_F32_16X16X4_F32 against a ones matrix ---------------
__device__ __forceinline__ float wave_sum32(float p) {
#ifdef USE_WMMA
  v2f a; a[0] = p; a[1] = 0.0f;       // A 16x4: lane partial at K=0 (or K=2)
  v2f b; b[0] = 1.0f; b[1] = 1.0f;    // B 4x16 all-ones
  v8f c = {};
  c = __builtin_amdgcn_wmma_f32_16x16x4_f32(false, a, false, b, (short)0, c,
                                            false, false);
  // D[m][n] = p_m + p_{m+16}; lane holds one column: sum its 8 rows,
  // then add the other half-column (lanes 0-15 hold M=0..7, 16-31 hold M=8..15)
  float s = c[0] + c[1] + c[2] + c[3] + c[4] + c[5] + c[6] + c[7];
  s += __shfl_xor(s, 16, 32);
  return s;
#else
  for (int d = 16; d > 0; d >>= 1) p += __shfl_xor(p, d, 32);
  return p;
#endif
}

// ---- wave-parallel searchsorted over LDS cumsum: 3 dependent LDS loads ------
// lower: first i with cs[i] >= t ; upper: first i with cs[i] > t
__device__ __forceinline__ int lower_bound_wave(const float* cs, float t, int lane) {
  const float c1 = cs[lane * 64 + 63];                  // chunk max
  const int chunk = __popc((unsigned)__ballot(c1 < t)); // prefix of ones
  if (chunk == 32) return BINS;                         // uniform branch
  const float c2 = cs[chunk * 64 + lane * 2 + 1];       // pair max
  const int pair = __popc((unsigned)__ballot(c2 < t));
  const int base = chunk * 64 + pair * 2;
  return base + ((cs[base] < t) ? 1 : 0);
}
__device__ __forceinline__ int upper_bound_wave(const float* cs, float t, int lane) {
  const float c1 = cs[lane * 64 + 63];
  const int chunk = __popc((unsigned)__ballot(c1 <= t));
  if (chunk == 32) return BINS;
  const float c2 = cs[chunk * 64 + lane * 2 + 1];
  const int pair = __popc((unsigned)__ballot(c2 <= t));
  const int base = chunk * 64 + pair * 2;
  return base + ((cs[base] <= t) ? 1 : 0);
}

// ---- pass 3: sequential greedy search, one wave32 ---------------------------
__global__ void __launch_bounds__(32) search_kernel(const float* __restrict__ hist,
                                                    const float* __restrict__ mm,
                                                    float* __restrict__ out) {
  __shared__ float cs[BINS];
  const int lane = threadIdx.x;
  const float min_val = mm[0];
  const float max_val = mm[1];
  const float bin_width = (max_val - min_val) / (float)BINS;

  float h[64];  // histogram pinned in VGPRs: bin i = j*32 + lane (static idx only)
#pragma unroll
  for (int j = 0; j < 64; ++j) h[j] = hist[j * 32 + lane];

  // inclusive cumsum into LDS (wave scan, 64 chunks of 32); reads hist directly
  // so this loop can stay rolled without forcing h[] into scratch.
  float run = 0.0f;
  for (int j = 0; j < 64; ++j) {
    float s = hist[j * 32 + lane];
#pragma unroll
    for (int d = 1; d < 32; d <<= 1) {
      float tt = __shfl_up(s, d, 32);
      if (lane >= d) s += tt;
    }
    cs[j * 32 + lane] = run + s;
    run += __shfl(s, 31, 32);
  }
  const float total = run;
  __syncthreads();  // single-wave WG: lowers to waits/S_NOP

  float alpha = 0.0f, beta = 1.0f;
  int sb = 0, eb = BINS - 1;
  float nm = __builtin_inff();
  bool done = false;
  const float inv_bw = (bin_width != 0.0f) ? (1.0f / bin_width) : 0.0f;

  while ((alpha < beta) && !done) {
    const float na = alpha + STEPF;
    const float nb = beta - STEPF;

    int l = lower_bound_wave(cs, na * total, lane);
    int r = upper_bound_wave(cs, nb * total, lane) - 1;
    l = min(max(l, sb), eb);
    r = min(max(r, sb), eb);

    const bool take_left = (l - sb) > (eb - r);
    const int nsb = take_left ? l : sb;
    const int neb = take_left ? eb : r;
    if (take_left) alpha = na; else beta = nb;

    const bool changed = (nsb != sb) || (neb != eb);
    if (changed) {  // uniform; EXEC stays all-ones for the WMMA below
      const float dst_w = bin_width * (float)(neb - nsb + 1) / (float)DST_NBINS;
      const float safe_w = (dst_w == 0.0f) ? 1.0f : dst_w;
      const float inv_w = 1.0f / safe_w;
      const float hw = 0.5f * safe_w;
      const float hw3 = hw * hw * hw;
      float acc0 = 0.0f, acc1 = 0.0f, acc2 = 0.0f, acc3 = 0.0f;
#pragma unroll
      for (int j = 0; j < 64; ++j) {   // FULL unroll: h[j] stays in VGPRs
        const int i = j * 32 + lane;
        const float begin = (float)(i - nsb) * bin_width;
        const float end = begin + bin_width;
        const float db = fminf(fmaxf(floorf(begin * inv_w), 0.0f), (float)(DST_NBINS - 1));
        const float de = fminf(fmaxf(floorf(end * inv_w), 0.0f), (float)(DST_NBINS - 1));
        const float density = h[j] * inv_bw;
        const float a0 = begin - (db + 0.5f) * safe_w;
        const float b1 = end - (de * safe_w + hw);
        const float t0 = hw3 - a0 * a0 * a0;                 // left partial dst bin
        const float tm = (de - db - 1.0f) * (2.0f * hw3);    // full middle dst bins
        const float t2 = b1 * b1 * b1 + hw3;                 // right partial dst bin
        const float term = density * (t0 + tm + t2) * (1.0f / 3.0f);
        if ((j & 3) == 0) acc0 += term;
        else if ((j & 3) == 1) acc1 += term;
        else if ((j & 3) == 2) acc2 += term;
        else acc3 += term;
      }
      float norm = wave_sum32((acc0 + acc1) + (acc2 + acc3));
      norm = (dst_w == 0.0f) ? 0.0f : norm;
      const bool broke = (norm > nm);
      if (!broke) { sb = nsb; eb = neb; nm = norm; }
      done = broke;
    }
  }

  if (lane == 0) {
    out[0] = min_val + bin_width * (float)sb;
    out[1] = min_val + bin_width * (float)(eb + 1);
  }
}

// ---- entry ------------------------------------------------------------------
extern "C" void kernel_launch(void* const* d_in, const int* in_sizes, int n_in,
                              void* d_out, int out_size, void* d_ws, size_t ws_size,
                              hipStream_t stream) {
  (void)n_in; (void)out_size; (void)ws_size;
  const float* x = (const float*)d_in[0];
  const int n = in_sizes[0];
  float* ws = (float*)d_ws;  // [0]=min [1]=max [2..2049]=hist

  init_kernel<<<8, 256, 0, stream>>>(ws);
  minmax_kernel<<<1024, 256, 0, stream>>>(x, n, ws);
  hist_kernel<<<1024, 256, 0, stream>>>(x, n, ws, ws + 2);
  search_kernel<<<1, 32, 0, stream>>>(ws + 2, ws, (float*)d_out);
}